// Router_68161130988043
// MI455X (gfx1250) — compile-verified
//
#include <hip/hip_runtime.h>

typedef float v2f __attribute__((ext_vector_type(2)));
typedef float v8f __attribute__((ext_vector_type(8)));

#define N_TOK   4096
#define N_CH    64
#define N_EXP   16
#define SPATIAL 256
#define CCAP    384
#define OUTMAT  (4096UL * 16UL * 384UL)   /* 25,165,824 elements per dense output */

// ---------------------------------------------------------------------------
// Zero-fill the two dense outputs (201 MB) with streaming b128 stores.
// ---------------------------------------------------------------------------
__global__ __launch_bounds__(256) void zero_kernel(float4* __restrict__ out, long n4) {
    long i      = (long)blockIdx.x * blockDim.x + threadIdx.x;
    long stride = (long)gridDim.x * blockDim.x;
    float4 z; z.x = 0.f; z.y = 0.f; z.z = 0.f; z.w = 0.f;
    for (; i < n4; i += stride) out[i] = z;
}

// ---------------------------------------------------------------------------
// Global average pool: one wave per (token, channel). 32 lanes x float4 pairs
// cover the 256 contiguous floats; shuffle tree reduces within the wave.
// pooled is [N_TOK][N_CH] flat, and X's (n,c) chunks are contiguous, so the
// global wave id IS the pooled index.
// ---------------------------------------------------------------------------
__global__ __launch_bounds__(256) void pool_kernel(const float* __restrict__ X,
                                                   float* __restrict__ pooled) {
    int wave = blockIdx.x * (blockDim.x >> 5) + (threadIdx.x >> 5);
    int lane = threadIdx.x & 31;
    const float4* p = (const float4*)(X + (size_t)wave * SPATIAL);
    float4 a = p[lane];
    float4 b = p[lane + 32];
    float s = (a.x + a.y + a.z + a.w) + (b.x + b.y + b.z + b.w);
    #pragma unroll
    for (int off = 16; off > 0; off >>= 1) s += __shfl_down(s, off, 32);
    if (lane == 0) pooled[wave] = s * (1.0f / 256.0f);
}

// ---------------------------------------------------------------------------
// Gating GEMM via V_WMMA_F32_16X16X4_F32: [4096x64] @ [64x16] + bias.
// One wave = one 16-token tile, 16 chained WMMAs over K=64.
// A layout (16x4 f32): lanes 0-15 -> K = k,k+1 ; lanes 16-31 -> K = k+2,k+3.
// B layout (4x16 f32): mirrored; N = lane&15.
// C layout (16x16 f32): VGPR i -> row i (lanes 0-15) / row i+8 (lanes 16-31).
// Grid sized exactly (no divergence): EXEC all-ones as WMMA requires.
// ---------------------------------------------------------------------------
__global__ __launch_bounds__(256) void gemm_logits_kernel(const float* __restrict__ pooled,
                                                          const float* __restrict__ Wg,
                                                          const float* __restrict__ bg,
                                                          float* __restrict__ logits) {
    int wave = blockIdx.x * (blockDim.x >> 5) + (threadIdx.x >> 5);
    int lane = threadIdx.x & 31;
    int tok0 = wave * 16;
    int m    = lane & 15;
    int kh   = (lane >> 4) * 2;   // 0 for lanes 0-15, 2 for lanes 16-31

    v8f acc = {};
    #pragma unroll
    for (int k = 0; k < N_CH; k += 4) {
        int c = k + kh;
        v2f a, b;
        a.x = pooled[(size_t)(tok0 + m) * N_CH + c];
        a.y = pooled[(size_t)(tok0 + m) * N_CH + c + 1];
        b.x = Wg[c * N_EXP + m];
        b.y = Wg[(c + 1) * N_EXP + m];
        acc = __builtin_amdgcn_wmma_f32_16x16x4_f32(false, a, false, b,
                                                    (short)0, acc, false, false);
    }

    float bias = bg[m];
    #pragma unroll
    for (int i = 0; i < 8; ++i) {
        int row = i + (lane >> 4) * 8;
        logits[(size_t)(tok0 + row) * N_EXP + m] = acc[i] + bias;
    }
}

// ---------------------------------------------------------------------------
// Single-block finalize: softmax/argmax per token, mean-LSE (z_loss),
// density-based aux_loss, token-ORDERED per-expert prefix scan for the
// capacity positions, then sparse scatter of dispatch/combine nonzeros.
// Thread t owns tokens [16t, 16t+16) -> global token order is preserved by
// (thread order, in-thread order), so the LDS histogram + exclusive scan
// reproduces jnp.cumsum exactly.
// ---------------------------------------------------------------------------
__global__ __launch_bounds__(256) void finalize_kernel(const float* __restrict__ logits,
                                                       float* __restrict__ out) {
    const int t = threadIdx.x;
    __shared__ int   hist[256 * N_EXP];   // per-thread per-expert counts -> exclusive prefix
    __shared__ float ps[256 * N_EXP];     // per-thread per-expert prob sums
    __shared__ float lsebuf[256];
    __shared__ float dens1[N_EXP];
    __shared__ float dens2[N_EXP];

    #pragma unroll
    for (int e = 0; e < N_EXP; ++e) hist[t * N_EXP + e] = 0;

    float psloc[N_EXP];
    #pragma unroll
    for (int e = 0; e < N_EXP; ++e) psloc[e] = 0.f;
    float lse_acc = 0.f;
    int   eidx[16];
    float gval[16];

    const int tok0 = t * 16;
    #pragma unroll
    for (int j = 0; j < 16; ++j) {
        const float* lg = logits + (size_t)(tok0 + j) * N_EXP;
        float l[N_EXP];
        #pragma unroll
        for (int e = 0; e < N_EXP; ++e) l[e] = lg[e];

        float mx = l[0]; int am = 0;
        #pragma unroll
        for (int e = 1; e < N_EXP; ++e) if (l[e] > mx) { mx = l[e]; am = e; }

        float pe[N_EXP]; float s = 0.f;
        #pragma unroll
        for (int e = 0; e < N_EXP; ++e) { pe[e] = __expf(l[e] - mx); s += pe[e]; }
        float inv = 1.f / s;
        #pragma unroll
        for (int e = 0; e < N_EXP; ++e) psloc[e] += pe[e] * inv;

        lse_acc += mx + __logf(s);
        hist[t * N_EXP + am] += 1;      // own row, no conflict
        eidx[j] = am;
        gval[j] = inv;                  // top-1 prob = exp(0)/s
    }

    #pragma unroll
    for (int e = 0; e < N_EXP; ++e) ps[t * N_EXP + e] = psloc[e];
    lsebuf[t] = lse_acc;
    __syncthreads();

    // Exclusive scan across threads (thread e handles expert e) + densities.
    if (t < N_EXP) {
        int run = 0;
        for (int i = 0; i < 256; ++i) {
            int v = hist[i * N_EXP + t];
            hist[i * N_EXP + t] = run;
            run += v;
        }
        dens1[t] = (float)run * (1.0f / 4096.0f);
        float sp = 0.f;
        for (int i = 0; i < 256; ++i) sp += ps[i * N_EXP + t];
        dens2[t] = sp * (1.0f / 4096.0f);
    }
    __syncthreads();

    if (t == 0) {
        float z = 0.f;
        for (int i = 0; i < 256; ++i) z += lsebuf[i];
        float aux = 0.f;
        for (int e = 0; e < N_EXP; ++e) aux += dens1[e] * dens2[e];
        out[2 * OUTMAT]     = z * (1.0f / 4096.0f);   // z_loss
        out[2 * OUTMAT + 1] = aux * (float)N_EXP;     // aux_loss
    }

    // Scatter nonzeros: position = exclusive prefix + in-thread running count.
    #pragma unroll
    for (int j = 0; j < 16; ++j) {
        int e = eidx[j];
        int p = hist[t * N_EXP + e];
        hist[t * N_EXP + e] = p + 1;
        if (p < CCAP) {
            size_t idx = ((size_t)(tok0 + j) * N_EXP + (size_t)e) * (size_t)CCAP + (size_t)p;
            out[idx]          = 1.0f;      // dispatch
            out[OUTMAT + idx] = gval[j];   // combine
        }
    }
}

// ---------------------------------------------------------------------------
extern "C" void kernel_launch(void* const* d_in, const int* in_sizes, int n_in,
                              void* d_out, int out_size, void* d_ws, size_t ws_size,
                              hipStream_t stream) {
    const float* X  = (const float*)d_in[0];   // [4096,64,16,16]
    const float* Wg = (const float*)d_in[1];   // [64,16]
    const float* bg = (const float*)d_in[2];   // [16]
    float* out = (float*)d_out;

    float* pooled = (float*)d_ws;                       // 4096*64 floats (1 MB)
    float* logits = pooled + (size_t)N_TOK * N_CH;      // 4096*16 floats (256 KB)

    // 2*OUTMAT floats are float4-aligned (OUTMAT % 4 == 0); the 2 trailing
    // scalars are written unconditionally by finalize_kernel.
    zero_kernel<<<2048, 256, 0, stream>>>((float4*)out, (long)(2UL * OUTMAT / 4UL));

    // 4096 tokens * 64 channels = 262144 waves, 8 waves per block.
    pool_kernel<<<N_TOK * N_CH / 8, 256, 0, stream>>>(X, pooled);

    // 4096/16 = 256 wave-tiles, 8 waves per block -> 32 blocks (exact, no tail).
    gemm_logits_kernel<<<32, 256, 0, stream>>>(pooled, Wg, bg, logits);

    finalize_kernel<<<1, 256, 0, stream>>>(logits, out);
}